// SUBGRAPH_5231270167316
// MI455X (gfx1250) — compile-verified
//
#include <hip/hip_runtime.h>

typedef __attribute__((ext_vector_type(2))) float v2f;
typedef __attribute__((ext_vector_type(8))) float v8f;

// One block (256 thr = 8 waves) per batch. Wave w computes the 16 dot products
// for edges [w*16, w*16+16) of this batch's 128 live rows (i in {0,1}, j in [0,64)),
// using V_WMMA_F32_16X16X4_F32 with B = W broadcast across all 16 N columns.
__global__ void __launch_bounds__(256)
score_wmma_kernel(const float* __restrict__ s_e,
                  const float* __restrict__ adj,
                  const float* __restrict__ W,
                  const float* __restrict__ bias,
                  float* __restrict__ out_score /* d_out + 64, (32,2,64) */) {
  __shared__ float ldsW[512];
  const int batch = blockIdx.x;
  const int tid   = threadIdx.x;            // 0..255

  // Stage W (512 f32 = 2 KB) into LDS once per block.
  ldsW[tid]       = W[tid];
  ldsW[tid + 256] = W[tid + 256];
  __syncthreads();

  const int wave = tid >> 5;                // 0..7
  const int lane = tid & 31;
  const int m    = lane & 15;               // row-within-tile this lane feeds A for
  const int hi   = lane >> 4;               // 0: K pair {0,1}; 1: K pair {2,3}

  const int edge = wave * 16 + m;           // 0..127 within batch
  const int i    = edge >> 6;               // 0 or 1 (only live rows)
  const int j    = edge & 63;
  const float* rowp =
      s_e + ((((size_t)batch * 64 + i) * 64) + j) * 512;

  v8f acc = {};
  #pragma unroll 4
  for (int k = 0; k < 512; k += 4) {
    const int kk = k + 2 * hi;
    v2f a = *(const v2f*)(rowp + kk);       // A: row m, K = kk, kk+1
    v2f b = *(const v2f*)(&ldsW[kk]);       // B: W[kk], W[kk+1] (all N cols equal)
    // 8 args: (neg_a, A, neg_b, B, c_mod, C, reuse_a, reuse_b)
    acc = __builtin_amdgcn_wmma_f32_16x16x4_f32(
        false, a, false, b, (short)0, acc, false, false);
  }

  const float b0 = bias[0];
  // D layout: VGPR r -> row r (lanes 0-15) / row 8+r (lanes 16-31); all N equal.
  if (m == 0) {
    #pragma unroll
    for (int r = 0; r < 8; ++r) {
      const int e  = wave * 16 + hi * 8 + r;
      const int ii = e >> 6;
      const int jj = e & 63;
      float av = adj[((size_t)batch * 64 + ii) * 64 + jj];
      if ((ii == 0 && jj == 1) || (ii == 1 && jj == 0)) av = 0.0f;  // adj edit
      out_score[(size_t)batch * 128 + e] = (acc[r] + b0) * av;
    }
  }
}

// One wave: lane b handles batch b. First-occurrence argmax (strict >) matches
// jnp.argmax; then the 0/1/2/3 flag.
__global__ void __launch_bounds__(32)
argmax_flag_kernel(const float* __restrict__ score,   /* d_out + 64   */
                   float* __restrict__ final_id,      /* d_out        */
                   float* __restrict__ flag) {        /* d_out + 4160 */
  const int b = threadIdx.x;
  if (b >= 32) return;
  const float* s0 = score + (size_t)b * 128;
  const float* s1 = s0 + 64;
  float m0 = s0[0], m1 = s1[0];
  int i0 = 0, i1 = 0;
  for (int jj = 1; jj < 64; ++jj) {
    float v0 = s0[jj]; if (v0 > m0) { m0 = v0; i0 = jj; }
    float v1 = s1[jj]; if (v1 > m1) { m1 = v1; i1 = jj; }
  }
  final_id[b * 2 + 0] = (float)i0;
  final_id[b * 2 + 1] = (float)i1;
  const bool sp = i0 > 0, op = i1 > 0;
  flag[b] = (sp && op) ? 3.0f : (sp ? 1.0f : (op ? 2.0f : 0.0f));
}

extern "C" void kernel_launch(void* const* d_in, const int* in_sizes, int n_in,
                              void* d_out, int out_size, void* d_ws, size_t ws_size,
                              hipStream_t stream) {
  const float* s_e  = (const float*)d_in[0];   // (32,64,64,512) f32
  const float* adj  = (const float*)d_in[1];   // (32,64,64)     f32
  const float* W    = (const float*)d_in[2];   // (512,1)        f32
  const float* bias = (const float*)d_in[3];   // (1,)           f32

  float* out      = (float*)d_out;
  float* final_id = out;            // 64 elems
  float* score    = out + 64;       // 4096 elems (32,2,64)
  float* flag     = out + 64 + 4096;// 32 elems

  score_wmma_kernel<<<32, 256, 0, stream>>>(s_e, adj, W, bias, score);
  argmax_flag_kernel<<<1, 32, 0, stream>>>(score, final_id, flag);
}